// LatentSpace_35588099015495
// MI455X (gfx1250) — compile-verified
//
#include <hip/hip_runtime.h>
#include <hip/hip_bf16.h>
#include <math.h>

typedef __attribute__((ext_vector_type(2))) float v2f;
typedef __attribute__((ext_vector_type(8))) float v8f;

#define N_STATES 64
#define DIM      16
#define NPOS     4096        // BATCH * L
#define N_CAT    2080        // 64*65/2
#define LOG2PI   1.8377869f  // log(2*pi) fp32
#define LOG_EPS  -36.841362f // log(1e-16)
#define RSQRT2   0.70710678f

// ---------------------------------------------------------------------------
// 1) logsumexp(log_pi) -> lse[0]   (one block, fixed-order tree: deterministic)
// ---------------------------------------------------------------------------
__global__ void lse_kernel(const float* __restrict__ logpi, float* __restrict__ lse) {
    __shared__ float sm[256];
    int t = threadIdx.x;
    float m = -INFINITY;
    for (int c = t; c < N_CAT; c += 256) m = fmaxf(m, logpi[c]);
    sm[t] = m; __syncthreads();
    for (int off = 128; off > 0; off >>= 1) {
        if (t < off) sm[t] = fmaxf(sm[t], sm[t + off]);
        __syncthreads();
    }
    float mx = sm[0]; __syncthreads();
    float s = 0.0f;
    for (int c = t; c < N_CAT; c += 256) s += expf(logpi[c] - mx);
    sm[t] = s; __syncthreads();
    for (int off = 128; off > 0; off >>= 1) {
        if (t < off) sm[t] += sm[t + off];
        __syncthreads();
    }
    if (t == 0) lse[0] = mx + logf(sm[0]);
}

// ---------------------------------------------------------------------------
// 2) Per-category constants.
//    catA = {c_alpha, c_bb, recip, inv_sig}, catB = {sq, base, lp, diagflag}
//    cab  = {a_idx, b_idx}
// ---------------------------------------------------------------------------
__global__ void cat_kernel(const float* __restrict__ mu,     // (16,64) row-major
                           const float* __restrict__ logpi,
                           const float* __restrict__ lse,
                           float4* __restrict__ catA,
                           float4* __restrict__ catB,
                           int2*   __restrict__ cab) {
    int c = blockIdx.x * blockDim.x + threadIdx.x;
    if (c >= N_CAT) return;
    // Row-major enumeration of upper triangle: for a: for b>=a
    int a = 0, rem = c;
    while (rem >= (N_STATES - a)) { rem -= (N_STATES - a); ++a; }
    int b = a + rem;

    float inv_sig = 0.0f, c_alpha = 0.0f, c_bb = 0.0f;
#pragma unroll
    for (int d = 0; d < DIM; ++d) {
        float mA = mu[d * N_STATES + a];
        float mB = mu[d * N_STATES + b];
        float al = mB - mA;
        inv_sig = fmaf(al, al, inv_sig);
        c_alpha = fmaf(al, mB, c_alpha);
        c_bb    = fmaf(mB, mB, c_bb);
    }
    float recip = (inv_sig == 0.0f) ? 0.0f : (1.0f / inv_sig);
    float cl = fminf(fmaxf(inv_sig, 1e-12f), 1e+30f);
    float sq = sqrtf(cl);
    // log(clip(softmax,eps,1)) = clamp(log_pi - lse, log(eps), 0)
    float lp = fminf(fmaxf(logpi[c] - lse[0], LOG_EPS), 0.0f);
    float base = lp + 0.5f * (LOG2PI - logf(cl));
    float diag = (inv_sig == 0.0f) ? 1.0f : 0.0f;
    catA[c] = make_float4(c_alpha, c_bb, recip, inv_sig);
    catB[c] = make_float4(sq, base, lp, diag);
    cab[c]  = make_int2(a, b);
}

// ---------------------------------------------------------------------------
// 3) ||z_bl||^2 per position
// ---------------------------------------------------------------------------
__global__ void z2_kernel(const float* __restrict__ z, float* __restrict__ z2) {
    int p = blockIdx.x * blockDim.x + threadIdx.x;
    if (p >= NPOS) return;
    const float4* zr = reinterpret_cast<const float4*>(z + p * DIM);
    float s = 0.0f;
#pragma unroll
    for (int q = 0; q < 4; ++q) {
        float4 v = zr[q];
        s += v.x * v.x + v.y * v.y + v.z * v.z + v.w * v.w;
    }
    z2[p] = s;
}

// ---------------------------------------------------------------------------
// 4) S(4096x64) = Z(4096x16) * mu(16x64) via V_WMMA_F32_16X16X4_F32 (fp32).
//    8 waves/block, 1 (16x16) output tile per wave, K=16 as 4 chained K=4 steps.
// ---------------------------------------------------------------------------
__global__ __launch_bounds__(256) void gemm_kernel(const float* __restrict__ z,
                                                   const float* __restrict__ mu,
                                                   float* __restrict__ S) {
    int lane = threadIdx.x & 31;
    int wave = threadIdx.x >> 5;
    int tile = blockIdx.x * 8 + wave;     // 0..1023  (256 M-tiles x 4 N-tiles)
    int m0 = (tile >> 2) * 16;
    int n0 = (tile & 3)  * 16;
    int half = lane >> 4;                 // 0: K even pair / M 0..7 ; 1: K odd pair / M 8..15
    int l15  = lane & 15;
    int arow = m0 + l15;                  // A: M = lane%16
    int bcol = n0 + l15;                  // B/D: N = lane%16

    v8f acc = {};
#pragma unroll
    for (int kc = 0; kc < 4; ++kc) {
        int k0 = kc * 4 + half * 2;       // lanes 0-15 hold K={k,k+1}, lanes 16-31 K={k+2,k+3}
        v2f a, bfrag;
        a.x = z[arow * DIM + k0];
        a.y = z[arow * DIM + k0 + 1];
        bfrag.x = mu[(k0)     * N_STATES + bcol];
        bfrag.y = mu[(k0 + 1) * N_STATES + bcol];
        acc = __builtin_amdgcn_wmma_f32_16x16x4_f32(
            /*neg_a=*/false, a, /*neg_b=*/false, bfrag,
            /*c_mod=*/(short)0, acc, /*reuse_a=*/false, /*reuse_b=*/false);
    }
#pragma unroll
    for (int i = 0; i < 8; ++i) {
        int r = m0 + i + half * 8;        // C/D: VGPR i -> rows i (lanes 0-15) / i+8 (lanes 16-31)
        S[r * N_STATES + bcol] = acc[i];
    }
}

// ---------------------------------------------------------------------------
// 5) Main pass: one wave per position; 65 categories per lane; online LSE.
// ---------------------------------------------------------------------------
__global__ __launch_bounds__(256) void main_kernel(const float*  __restrict__ S,
                                                   const float*  __restrict__ z2,
                                                   const float4* __restrict__ catA,
                                                   const float4* __restrict__ catB,
                                                   const int2*   __restrict__ cab,
                                                   float* __restrict__ perpos) {
    __shared__ float srow[8][N_STATES];
    int lane = threadIdx.x & 31;
    int wave = threadIdx.x >> 5;
    int pos = blockIdx.x * 8 + wave;      // 512 blocks * 8 waves = 4096
    srow[wave][lane]      = S[pos * N_STATES + lane];
    srow[wave][lane + 32] = S[pos * N_STATES + lane + 32];
    __syncthreads();
    float z2v = z2[pos];

    float m = -INFINITY, s = 0.0f;
    // 2080 = 65 * 32: every lane does exactly 65 categories, no divergence.
    for (int i = 0; i < 65; ++i) {
        int c = i * 32 + lane;
        float4 A = catA[c];
        float4 B = catB[c];
        int2  ab = cab[c];
        float ga = srow[wave][ab.x];
        float gb = srow[wave][ab.y];
        float dotv = gb - ga - A.x;                        // sum_d alpha*beta
        float bb   = z2v - 2.0f * gb + A.y;                // sum_d beta^2
        float nu   = -dotv * A.z;
        float t    = fmaf(nu * nu, A.w, -bb);              // -bb + nu^2*inv_sig
        float eta1 = (1.0f - nu) * B.x;
        float eta2 = -nu * B.x;
        float P = 0.5f * (erff(eta1 * RSQRT2) - erff(eta2 * RSQRT2));
        P = fmaxf(P, 1e-16f);
        float reg  = B.y + 0.5f * t + __logf(P);           // base + 0.5*t + log(P)
        float dia  = B.z - 0.5f * bb;                      // lp - 0.5*bb
        float x = (B.w != 0.0f) ? dia : reg;
        // online logsumexp
        if (x > m) { s = fmaf(s, __expf(m - x), 1.0f); m = x; }
        else       { s += __expf(x - m); }
    }
    // wave32 shuffle reduction of (m, s)
    for (int off = 16; off > 0; off >>= 1) {
        float mo = __shfl_xor(m, off, 32);
        float so = __shfl_xor(s, off, 32);
        float mn = fmaxf(m, mo);
        s = s * __expf(m - mn) + so * __expf(mo - mn);
        m = mn;
    }
    if (lane == 0)
        perpos[pos] = m + logf(s) - 0.5f * DIM * LOG2PI;
}

// ---------------------------------------------------------------------------
// 6) Deterministic mean over 4096 positions
// ---------------------------------------------------------------------------
__global__ void mean_kernel(const float* __restrict__ perpos, float* __restrict__ out) {
    __shared__ float sm[256];
    int t = threadIdx.x;
    float s = 0.0f;
    for (int i = 0; i < 16; ++i) s += perpos[t + i * 256];
    sm[t] = s; __syncthreads();
    for (int off = 128; off > 0; off >>= 1) {
        if (t < off) sm[t] += sm[t + off];
        __syncthreads();
    }
    if (t == 0) out[0] = sm[0] * (1.0f / (float)NPOS);
}

// ---------------------------------------------------------------------------
extern "C" void kernel_launch(void* const* d_in, const int* in_sizes, int n_in,
                              void* d_out, int out_size, void* d_ws, size_t ws_size,
                              hipStream_t stream) {
    const float* z     = (const float*)d_in[0];   // (256,16,16)
    const float* mu    = (const float*)d_in[1];   // (16,64)
    const float* logpi = (const float*)d_in[2];   // (1,2080)
    float* out = (float*)d_out;

    // workspace layout (all offsets 16B aligned)
    float*  S      = (float*)d_ws;                          // 4096*64
    float*  z2     = S + NPOS * N_STATES;                   // 4096
    float4* catA   = (float4*)(z2 + NPOS);                  // 2080
    float4* catB   = catA + N_CAT;                          // 2080
    int2*   cab    = (int2*)(catB + N_CAT);                 // 2080
    float*  perpos = (float*)(cab + N_CAT);                 // 4096
    float*  lse    = perpos + NPOS;                         // 1

    lse_kernel <<<1, 256, 0, stream>>>(logpi, lse);
    cat_kernel <<<(N_CAT + 255) / 256, 256, 0, stream>>>(mu, logpi, lse, catA, catB, cab);
    z2_kernel  <<<NPOS / 256, 256, 0, stream>>>(z, z2);
    gemm_kernel<<<128, 256, 0, stream>>>(z, mu, S);         // 1024 waves, 4 WMMAs each
    main_kernel<<<NPOS / 8, 256, 0, stream>>>(S, z2, catA, catB, cab, perpos);
    mean_kernel<<<1, 256, 0, stream>>>(perpos, out);
}